// AttributeDecoder_71657234366668
// MI455X (gfx1250) — compile-verified
//
#include <hip/hip_runtime.h>

typedef __attribute__((ext_vector_type(2))) float v2f;
typedef __attribute__((ext_vector_type(8))) float v8f;

static constexpr int Dd = 512;     // emb dim
static constexpr int Vv = 16;      // outputs per head
static constexpr int Ee = 64;      // heads
static constexpr int Nn = 2 * 32 * 32 * 32; // 65536 voxels
static constexpr int WROW = 516;   // padded LDS row stride (floats): 516%64==4 -> conflict-free b64 reads

// ---------------- binning kernels ----------------

__global__ void zero_counts_kernel(int* counts) {
    int t = threadIdx.x;
    if (t < Ee) counts[t] = 0;
}

__global__ void hist_kernel(const int* __restrict__ btg, const int* __restrict__ b2h,
                            int* __restrict__ hvox, int* __restrict__ counts) {
    int i = blockIdx.x * blockDim.x + threadIdx.x;
    if (i < Nn) {
        int h = b2h[btg[i]];
        hvox[i] = h;
        atomicAdd(&counts[h], 1);
    }
}

__global__ void scan_kernel(const int* __restrict__ counts, int* __restrict__ cursors) {
    if (threadIdx.x == 0 && blockIdx.x == 0) {
        int acc = 0;
        for (int e = 0; e < Ee; ++e) {
            cursors[e] = acc;
            acc += counts[e];
        }
    }
}

__global__ void scatter_kernel(const int* __restrict__ hvox, int* __restrict__ cursors,
                               int* __restrict__ sidx, int* __restrict__ shead) {
    int i = blockIdx.x * blockDim.x + threadIdx.x;
    if (i < Nn) {
        int h = hvox[i];
        int p = atomicAdd(&cursors[h], 1);
        sidx[p]  = i;
        shead[p] = h;
    }
}

// ---------------- routed WMMA GEMM ----------------
// Workgroup = 128 threads (4 waves). Each wave handles 16 head-sorted voxels.
// For each head present in the group's (contiguous, sorted) head range:
//   - stage W[h] (16 x 512 fp32, padded rows) + bias into LDS using
//     global_load_async_to_lds_b128 (ASYNCcnt-tracked, no VGPR round-trip)
//   - wave computes D(16x16) = A(16 voxels x 512) * B(512 x 16) via 128x v_wmma_f32_16x16x4_f32
//   - rows whose head matches write out[voxel, 0..15]
__global__ void __launch_bounds__(128)
decoder_gemm_kernel(const float* __restrict__ x,
                    const float* __restrict__ W_heads,
                    const float* __restrict__ b_heads,
                    const int* __restrict__ sidx,
                    const int* __restrict__ shead,
                    float* __restrict__ out) {
    __shared__ float Wl[Vv * WROW];   // padded W[h] tile
    __shared__ float bl[Vv];
    __shared__ int   lidx[64];
    __shared__ int   lhead[64];

    const int tid  = threadIdx.x;
    const int lane = tid & 31;
    const int wv   = tid >> 5;            // wave id 0..3
    const int base = blockIdx.x * 64;     // 64 sorted voxels per workgroup (Nn/64 = 1024 groups)

    if (tid < 64) {
        lidx[tid]  = sidx[base + tid];
        lhead[tid] = shead[base + tid];
    }
    __syncthreads();

    const int m    = lane & 15;           // matrix row owned by this lane (A & store phases)
    const int n    = lane & 15;           // output column owned by this lane (B/D layout)
    const int koff = (lane >> 4) * 2;     // lanes 16..31 hold the upper K-half of each K=4 step

    // Per-lane pointer to this lane's A-row (gathered voxel feature row), reused across heads.
    const float* xrow = x + (size_t)lidx[wv * 16 + m] * Dd;

    // Heads are sorted -> the group's distinct heads are the contiguous range [h0, h1]
    // (typically a single value). readfirstlane makes the loop + addressing scalar.
    const int h0 = __builtin_amdgcn_readfirstlane(lhead[0]);
    const int h1 = __builtin_amdgcn_readfirstlane(lhead[63]);

    for (int hh = h0; hh <= h1; ++hh) {
        // ---- stage W[hh] into LDS: 2048 16B async copies across 128 threads ----
        // GVS addressing: saddr = uniform W_heads base (SGPR pair), voff carries
        // head offset + element offset. LDS dest address in a VGPR (flat low bits
        // of a __shared__ pointer == LDS byte offset).
        #pragma unroll
        for (int it = 0; it < (Vv * Dd / 4) / 128; ++it) {
            int t = it * 128 + tid;                     // float4 index within W[hh]
            int r = t >> 7;                             // / (Dd/4) : output row 0..15
            int c = t & 127;                            // % (Dd/4) : float4 column
            unsigned lds_addr = (unsigned)(uintptr_t)(&Wl[r * WROW + 4 * c]);
            unsigned goff     = (unsigned)(((size_t)hh * Vv * Dd + (size_t)t * 4) * sizeof(float));
            asm volatile("global_load_async_to_lds_b128 %0, %1, %2 offset:0"
                         :
                         : "v"(lds_addr), "v"(goff), "s"(W_heads)
                         : "memory");
        }
        if (tid < Vv) bl[tid] = b_heads[hh * Vv + tid];
        asm volatile("s_wait_asynccnt 0x0" ::: "memory");
        __syncthreads();

        // wave-uniform predicate: does this wave's tile contain any voxel routed to hh?
        const bool mine = (lhead[wv * 16 + m] == hh);
        if (__any(mine)) {
            v8f acc = {};
            #pragma unroll 4
            for (int k = 0; k < Dd; k += 4) {
                const int kk = k + koff;
                // A fragment: 16x4 fp32. lane L: M = L%16, VGPR0/1 = K = kk, kk+1 (contiguous b64 load)
                v2f a;
                a.x = xrow[kk];
                a.y = xrow[kk + 1];
                // B fragment: 4x16 fp32. lane L: N = L%16, VGPR0/1 = K = kk, kk+1 (conflict-free ds b64)
                v2f b;
                b.x = Wl[n * WROW + kk];
                b.y = Wl[n * WROW + kk + 1];
                acc = __builtin_amdgcn_wmma_f32_16x16x4_f32(
                    /*neg_a=*/false, a, /*neg_b=*/false, b,
                    /*c_mod=*/(short)0, acc, /*reuse_a=*/false, /*reuse_b=*/false);
            }
            // D layout: acc[j] holds M = j + 8*(L/16), N = L%16
            const float bias  = bl[n];
            const int   mbase = (lane >> 4) * 8;
            #pragma unroll
            for (int j = 0; j < 8; ++j) {
                const int vslot = wv * 16 + mbase + j;
                if (lhead[vslot] == hh) {
                    out[(size_t)lidx[vslot] * Vv + n] = acc[j] + bias;
                }
            }
        }
        __syncthreads();  // all waves done reading Wl/bl before next head's async staging
    }
}

// ---------------- launch ----------------
extern "C" void kernel_launch(void* const* d_in, const int* in_sizes, int n_in,
                              void* d_out, int out_size, void* d_ws, size_t ws_size,
                              hipStream_t stream) {
    const int*   block_type_grid = (const int*)d_in[0];   // (B,W,H,L) int32
    const float* x               = (const float*)d_in[1]; // (N, 512) fp32
    const float* W_heads         = (const float*)d_in[2]; // (64, 16, 512) fp32
    const float* b_heads         = (const float*)d_in[3]; // (64, 16) fp32
    const int*   block2head      = (const int*)d_in[4];   // (256,) int32
    float*       out             = (float*)d_out;         // (N, 16) fp32

    // workspace layout (ints): [hvox: N][counts: E][cursors: E][sidx: N][shead: N]
    int* ws      = (int*)d_ws;
    int* hvox    = ws;
    int* counts  = ws + Nn;
    int* cursors = ws + Nn + Ee;
    int* sidx    = ws + Nn + 2 * Ee;
    int* shead   = ws + Nn + 2 * Ee + Nn;

    const int threads = 256;
    const int blocksN = (Nn + threads - 1) / threads;

    zero_counts_kernel<<<1, Ee, 0, stream>>>(counts);
    hist_kernel<<<blocksN, threads, 0, stream>>>(block_type_grid, block2head, hvox, counts);
    scan_kernel<<<1, 64, 0, stream>>>(counts, cursors);
    scatter_kernel<<<blocksN, threads, 0, stream>>>(hvox, cursors, sidx, shead);

    decoder_gemm_kernel<<<Nn / 64, 128, 0, stream>>>(x, W_heads, b_heads, sidx, shead, out);
}